// SpecAugment_2869038154066
// MI455X (gfx1250) — compile-verified
//
#include <hip/hip_runtime.h>
#include <cstdint>

// ---------------------------------------------------------------------------
// SpecAugment for x:(128,1,128,2048) f32.  Memory-bound: ~268MB HBM traffic
// (second read of x hits the 192MB L2), floor ~12us @ 23.3 TB/s.
// Threefry-2x32 reproduces jax.random exactly; WMMA f32 16x16x4 (B=ones) is
// used as the intra-wave reduction primitive for the per-sample mean.
// ---------------------------------------------------------------------------

#define BATCH      128
#define NFREQ      128
#define NTIME      2048
#define ELEMS_PER_B (NFREQ * NTIME)          // 262144 floats
#define F4_PER_B   (ELEMS_PER_B / 4)         // 65536 float4
#define F4_PER_ROW (NTIME / 4)               // 512 float4

typedef __attribute__((ext_vector_type(2))) float v2f;
typedef __attribute__((ext_vector_type(8))) float v8f;

// ----------------------------- Threefry-2x32 -------------------------------
__device__ __forceinline__ uint32_t rotl32(uint32_t v, int s) {
  return (v << s) | (v >> (32 - s));
}

__device__ __forceinline__ void threefry2x32(uint32_t k0, uint32_t k1,
                                             uint32_t x0, uint32_t x1,
                                             uint32_t& o0, uint32_t& o1) {
  const uint32_t ks0 = k0, ks1 = k1, ks2 = k0 ^ k1 ^ 0x1BD11BDAu;
  const int r0[4] = {13, 15, 26, 6};
  const int r1[4] = {17, 29, 16, 24};
  x0 += ks0; x1 += ks1;
#pragma unroll
  for (int i = 0; i < 4; ++i) { x0 += x1; x1 = rotl32(x1, r0[i]); x1 ^= x0; }
  x0 += ks1; x1 += ks2 + 1u;
#pragma unroll
  for (int i = 0; i < 4; ++i) { x0 += x1; x1 = rotl32(x1, r1[i]); x1 ^= x0; }
  x0 += ks2; x1 += ks0 + 2u;
#pragma unroll
  for (int i = 0; i < 4; ++i) { x0 += x1; x1 = rotl32(x1, r0[i]); x1 ^= x0; }
  x0 += ks0; x1 += ks1 + 3u;
#pragma unroll
  for (int i = 0; i < 4; ++i) { x0 += x1; x1 = rotl32(x1, r1[i]); x1 ^= x0; }
  x0 += ks1; x1 += ks2 + 4u;
#pragma unroll
  for (int i = 0; i < 4; ++i) { x0 += x1; x1 = rotl32(x1, r0[i]); x1 ^= x0; }
  x0 += ks2; x1 += ks0 + 5u;
  o0 = x0; o1 = x1;
}

// split(key) with counts [0,1,2,3]:
//   kA = (H(0,2).w0, H(1,3).w0), kB = (H(0,2).w1, H(1,3).w1)
__device__ __forceinline__ void tf_split(uint32_t k0, uint32_t k1,
                                         uint32_t& a0, uint32_t& a1,
                                         uint32_t& b0, uint32_t& b1) {
  uint32_t h0a, h0b, h1a, h1b;
  threefry2x32(k0, k1, 0u, 2u, h0a, h0b);
  threefry2x32(k0, k1, 1u, 3u, h1a, h1b);
  a0 = h0a; a1 = h1a;   // first subkey
  b0 = h0b; b1 = h1b;   // second subkey
}

// Reproduces _axis_mask(key, 128, 1, axis_len, mask_param) for sample b.
__device__ __forceinline__ void axis_params(uint32_t k0, uint32_t k1, int b,
                                            int axis_len, int mask_param,
                                            int& start, int& width) {
  uint32_t kw0, kw1, ks0, ks1;
  tf_split(k0, k1, kw0, kw1, ks0, ks1);

  // randint(kw, (128,1), 0, span): bits shape (2,128,1) -> counts iota(256)
  //   higher = H(b, b+128).w0 ; lower = H(b, b+128).w1
  uint32_t hi, lo;
  threefry2x32(kw0, kw1, (uint32_t)b, (uint32_t)(b + 128), hi, lo);
  const int maxw = (mask_param < axis_len) ? mask_param : axis_len;
  const uint32_t span = (uint32_t)(maxw + 1);
  uint32_t mult = 65536u % span;
  mult = (mult * mult) % span;
  const uint32_t w = (hi * mult + lo) % span;   // uint32 wrap, then rem: as in JAX

  // uniform(ks, (128,1)): counts iota(128) -> pairs (i, i+64)
  uint32_t ub;
  if (b < 64) {
    uint32_t u0, u1;
    threefry2x32(ks0, ks1, (uint32_t)b, (uint32_t)(b + 64), u0, u1);
    ub = u0;
  } else {
    uint32_t u0, u1;
    threefry2x32(ks0, ks1, (uint32_t)(b - 64), (uint32_t)b, u0, u1);
    ub = u1;
  }
  float u = __uint_as_float((ub >> 9) | 0x3f800000u) - 1.0f;
  u = fmaxf(0.0f, u);
  const int st = (int)floorf(u * (float)(axis_len - (int)w + 1));
  start = st;
  width = (int)w;
}

// -------------------- Kernel 1: per-sample mask parameters -----------------
// mp[b] = {f_start, f_end, t_start, t_end}
__global__ void __launch_bounds__(128)
specaug_rng_kernel(int4* __restrict__ mp) {
  const int b = threadIdx.x;
  if (b >= BATCH) return;
  // root key(42) = (0,42); split -> kf, kt
  uint32_t kf0, kf1, kt0, kt1;
  tf_split(0u, 42u, kf0, kf1, kt0, kt1);

  int fs, fw, ts, tw;
  axis_params(kf0, kf1, b, NFREQ, 8, fs, fw);
  axis_params(kt0, kt1, b, NTIME, 16, ts, tw);
  mp[b] = make_int4(fs, fs + fw, ts, ts + tw);
}

// -------------------- Kernel 2: per-sample mean via WMMA -------------------
__global__ void __launch_bounds__(256)
specaug_mean_kernel(const float* __restrict__ x, float* __restrict__ means) {
  const int b = blockIdx.x;
  const float4* px = (const float4*)(x + (size_t)b * ELEMS_PER_B);

  float sum = 0.0f;
#pragma unroll 4
  for (int i = threadIdx.x; i < F4_PER_B; i += 256) {
    float4 v = px[i];
    sum += (v.x + v.y) + (v.z + v.w);
  }

  // Wave32 reduction with V_WMMA_F32_16X16X4_F32, B = ones:
  //   A[m,0]=partial[lane m], A[m,2]=partial[lane m+16], K=1,3 zero.
  //   D[m,n] = partial[m] + partial[m+16] (same for every column n).
  v2f a;  a[0] = sum;  a[1] = 0.0f;
  v2f bm; bm[0] = 1.0f; bm[1] = 1.0f;
  v8f c = {};
  v8f d = __builtin_amdgcn_wmma_f32_16x16x4_f32(
      /*neg_a=*/false, a, /*neg_b=*/false, bm,
      /*c_mod=*/(short)0, c, /*reuse_a=*/false, /*reuse_b=*/false);

  // lanes 0-15 hold rows 0-7, lanes 16-31 hold rows 8-15 -> xor-16 combine
  float part = ((d[0] + d[1]) + (d[2] + d[3])) + ((d[4] + d[5]) + (d[6] + d[7]));
  float wave_sum = part + __shfl_xor(part, 16, 32);

  __shared__ float lds[8];
  const int lane = threadIdx.x & 31;
  const int wave = threadIdx.x >> 5;
  if (lane == 0) lds[wave] = wave_sum;
  __syncthreads();
  if (threadIdx.x == 0) {
    float t = 0.0f;
#pragma unroll
    for (int i = 0; i < 8; ++i) t += lds[i];
    means[b] = t * (1.0f / (float)ELEMS_PER_B);
  }
}

// -------------------- Kernel 3: streaming masked fill ----------------------
__global__ void __launch_bounds__(256)
specaug_apply_kernel(const float* __restrict__ x, float* __restrict__ out,
                     const int4* __restrict__ mp, const float* __restrict__ means) {
  const size_t idx = (size_t)blockIdx.x * 256 + threadIdx.x;  // float4 index
  const int b   = (int)(idx >> 16);        // 65536 float4 per sample
  const int rem = (int)(idx & 65535u);
  const int f   = rem >> 9;                // 512 float4 per freq row
  const int t0  = (rem & 511) << 2;        // time index of lane's first elem

  const int4  m    = mp[b];
  const float fill = means[b];

  const float4 v = ((const float4*)x)[idx];
  const bool mf = (f >= m.x) & (f < m.y);

  float4 o;
  o.x = (mf | ((t0 + 0) >= m.z & (t0 + 0) < m.w)) ? fill : v.x;
  o.y = (mf | ((t0 + 1) >= m.z & (t0 + 1) < m.w)) ? fill : v.y;
  o.z = (mf | ((t0 + 2) >= m.z & (t0 + 2) < m.w)) ? fill : v.z;
  o.w = (mf | ((t0 + 3) >= m.z & (t0 + 3) < m.w)) ? fill : v.w;

  ((float4*)out)[idx] = o;
}

// ---------------------------------------------------------------------------
extern "C" void kernel_launch(void* const* d_in, const int* in_sizes, int n_in,
                              void* d_out, int out_size, void* d_ws, size_t ws_size,
                              hipStream_t stream) {
  (void)in_sizes; (void)n_in; (void)out_size; (void)ws_size;

  const float* x   = (const float*)d_in[0];
  float*       out = (float*)d_out;

  int4*  mp    = (int4*)d_ws;                                   // 128 * 16 B
  float* means = (float*)((char*)d_ws + BATCH * sizeof(int4));  // 128 * 4 B

  specaug_rng_kernel<<<1, 128, 0, stream>>>(mp);
  specaug_mean_kernel<<<BATCH, 256, 0, stream>>>(x, means);

  const int total_f4 = BATCH * F4_PER_B;                        // 8,388,608
  specaug_apply_kernel<<<total_f4 / 256, 256, 0, stream>>>(x, out, mp, means);
}